// GCN_Node_Classifier_49761491091779
// MI455X (gfx1250) — compile-verified
//
#include <hip/hip_runtime.h>
#include <hip/hip_bf16.h>

typedef __attribute__((ext_vector_type(2))) float v2f;
typedef __attribute__((ext_vector_type(8))) float v8f;

// ---------------------------------------------------------------------------
// Utility kernels
// ---------------------------------------------------------------------------
__global__ void k_zero(float* __restrict__ p, int n) {
    int t = blockIdx.x * blockDim.x + threadIdx.x;
    int stride = gridDim.x * blockDim.x;
    for (int i = t; i < n; i += stride) p[i] = 0.0f;
}

__global__ void k_degrees(const int* __restrict__ src, const int* __restrict__ dst,
                          float* __restrict__ deg_out, float* __restrict__ deg_in, int E) {
    int t = blockIdx.x * blockDim.x + threadIdx.x;
    if (t < E) {
        atomicAdd(&deg_out[src[t]], 1.0f);   // -> global_atomic_add_f32
        atomicAdd(&deg_in[dst[t]], 1.0f);
    }
}

__global__ void k_rsqrt(float* __restrict__ d, int n) {
    int t = blockIdx.x * blockDim.x + threadIdx.x;
    if (t < n) d[t] = rsqrtf(fmaxf(d[t], 1.0f));
}

// ---------------------------------------------------------------------------
// GEMM1: H1[N,64] = (X[N,128] * norm_src[:,None]) @ W1[128,64]
// block = 256 threads = 8 waves; block covers 32 rows x 64 cols
// wave (w) -> M-tile (w>>2)*16, N-tile (w&3)*16; K loop in steps of 4 via
// v_wmma_f32_16x16x4_f32 (f32 in / f32 acc; precision-preserving).
// ---------------------------------------------------------------------------
__global__ __launch_bounds__(256)
void k_gemm1(const float* __restrict__ X, const float* __restrict__ W,
             const float* __restrict__ norm_src, float* __restrict__ H, int N) {
    __shared__ float lA[32][130];   // padded: stride 130 breaks bank conflicts
    __shared__ float lB[128][66];   // W1 staged [k][n], padded stride 66

    const int m0 = blockIdx.x * 32;
    const int t  = threadIdx.x;

    // Stage A tile, fused with norm_src row scaling (clamped rows for tails)
    #pragma unroll
    for (int i = 0; i < 16; ++i) {
        int idx = t + i * 256;
        int r = idx >> 7, c = idx & 127;
        int row = m0 + r; if (row >= N) row = N - 1;
        lA[r][c] = X[(long)row * 128 + c] * norm_src[row];
    }
    // Stage W1 (128x64)
    #pragma unroll
    for (int i = 0; i < 32; ++i) {
        int idx = t + i * 256;
        lB[idx >> 6][idx & 63] = W[idx];
    }
    __syncthreads();

    const int wave = t >> 5, lane = t & 31;
    const int mt = (wave >> 2) * 16;   // 0 or 16
    const int nt = (wave & 3) * 16;    // 0..48
    const int lr = lane & 15;          // row/col within fragment
    const int lh = lane >> 4;          // half-wave selects K pair

    v8f acc = {};
    #pragma unroll
    for (int k0 = 0; k0 < 128; k0 += 4) {
        const int kk = k0 + 2 * lh;
        v2f a, b;
        a[0] = lA[mt + lr][kk];
        a[1] = lA[mt + lr][kk + 1];
        b[0] = lB[kk][nt + lr];
        b[1] = lB[kk + 1][nt + lr];
        acc = __builtin_amdgcn_wmma_f32_16x16x4_f32(
            /*neg_a=*/false, a, /*neg_b=*/false, b,
            /*c_mod=*/(short)0, acc, /*reuse_a=*/false, /*reuse_b=*/false);
    }

    // C/D layout: VGPR r -> M=r (lanes 0-15) / M=r+8 (lanes 16-31), N=lane&15
    #pragma unroll
    for (int r = 0; r < 8; ++r) {
        int m = m0 + mt + r + 8 * lh;
        if (m < N) H[(long)m * 64 + nt + lr] = acc[r];
    }
}

// ---------------------------------------------------------------------------
// Scatter-add layer 1: 16 lanes per edge, float4 gather, fp32 atomics (L2-resident)
// ---------------------------------------------------------------------------
__global__ void k_scatter1(const float* __restrict__ H, const int* __restrict__ src,
                           const int* __restrict__ dst, float* __restrict__ AGG, int E) {
    long t = (long)blockIdx.x * blockDim.x + threadIdx.x;
    int e = (int)(t >> 4);
    if (e >= E) return;
    int f = ((int)t & 15) * 4;
    int s = src[e], d = dst[e];
    const float4 v = *(const float4*)(H + (long)s * 64 + f);
    float* out = AGG + (long)d * 64 + f;
    atomicAdd(out + 0, v.x); atomicAdd(out + 1, v.y);
    atomicAdd(out + 2, v.z); atomicAdd(out + 3, v.w);
}

// ---------------------------------------------------------------------------
// GEMM2: H2[N,16] = relu(AGG1*norm_dst + b1) * norm_src @ W2[64,16]
// 8 waves x 16 rows = 128 rows per block; bias/ReLU/scales fused into A-stage.
// ---------------------------------------------------------------------------
__global__ __launch_bounds__(256)
void k_gemm2(const float* __restrict__ AGG1, const float* __restrict__ W2,
             const float* __restrict__ b1,
             const float* __restrict__ norm_src, const float* __restrict__ norm_dst,
             float* __restrict__ H2, int N) {
    __shared__ float lA[128][66];
    __shared__ float lB[64][18];

    const int m0 = blockIdx.x * 128;
    const int t  = threadIdx.x;

    #pragma unroll
    for (int i = 0; i < 32; ++i) {
        int idx = t + i * 256;
        int r = idx >> 6, c = idx & 63;
        int row = m0 + r; if (row >= N) row = N - 1;
        float v = AGG1[(long)row * 64 + c];
        v = fmaxf(v * norm_dst[row] + b1[c], 0.0f) * norm_src[row];
        lA[r][c] = v;
    }
    #pragma unroll
    for (int i = 0; i < 4; ++i) {
        int idx = t + i * 256;
        lB[idx >> 4][idx & 15] = W2[idx];
    }
    __syncthreads();

    const int wave = t >> 5, lane = t & 31;
    const int mt = wave * 16;
    const int lr = lane & 15;
    const int lh = lane >> 4;

    v8f acc = {};
    #pragma unroll
    for (int k0 = 0; k0 < 64; k0 += 4) {
        const int kk = k0 + 2 * lh;
        v2f a, b;
        a[0] = lA[mt + lr][kk];
        a[1] = lA[mt + lr][kk + 1];
        b[0] = lB[kk][lr];
        b[1] = lB[kk + 1][lr];
        acc = __builtin_amdgcn_wmma_f32_16x16x4_f32(
            false, a, false, b, (short)0, acc, false, false);
    }

    #pragma unroll
    for (int r = 0; r < 8; ++r) {
        int m = m0 + mt + r + 8 * lh;
        if (m < N) H2[(long)m * 16 + lr] = acc[r];
    }
}

// ---------------------------------------------------------------------------
// Scatter-add layer 2: 4 lanes per edge x float4
// ---------------------------------------------------------------------------
__global__ void k_scatter2(const float* __restrict__ H, const int* __restrict__ src,
                           const int* __restrict__ dst, float* __restrict__ AGG, int E) {
    long t = (long)blockIdx.x * blockDim.x + threadIdx.x;
    int e = (int)(t >> 2);
    if (e >= E) return;
    int f = ((int)t & 3) * 4;
    int s = src[e], d = dst[e];
    const float4 v = *(const float4*)(H + (long)s * 16 + f);
    float* out = AGG + (long)d * 16 + f;
    atomicAdd(out + 0, v.x); atomicAdd(out + 1, v.y);
    atomicAdd(out + 2, v.z); atomicAdd(out + 3, v.w);
}

__global__ void k_finalize(const float* __restrict__ AGG2, const float* __restrict__ norm_dst,
                           const float* __restrict__ b2, float* __restrict__ out, int total) {
    int t = blockIdx.x * blockDim.x + threadIdx.x;
    if (t < total) {
        int i = t >> 4, j = t & 15;
        out[t] = AGG2[t] * norm_dst[i] + b2[j];
    }
}

// ---------------------------------------------------------------------------
// Host-side launch sequence (graph-capture safe: stream-only, no allocs)
// ---------------------------------------------------------------------------
extern "C" void kernel_launch(void* const* d_in, const int* in_sizes, int n_in,
                              void* d_out, int out_size, void* d_ws, size_t ws_size,
                              hipStream_t stream) {
    const float* emb = (const float*)d_in[0];
    const int*   src = (const int*)  d_in[1];
    const int*   dst = (const int*)  d_in[2];
    const float* W1  = (const float*)d_in[3];
    const float* b1  = (const float*)d_in[4];
    const float* W2  = (const float*)d_in[5];
    const float* b2  = (const float*)d_in[6];
    float* out = (float*)d_out;

    const int N = in_sizes[0] / 128;
    const int E = in_sizes[1];

    // Workspace layout (floats). H2 aliases H1 region; AGG2 aliases AGG1 region.
    float* ws       = (float*)d_ws;
    float* norm_src = ws;                     // N
    float* norm_dst = ws + N;                 // N
    float* H1       = ws + 2 * (size_t)N;     // N*64  (reused as H2: N*16)
    float* AGG1     = H1 + (size_t)N * 64;    // N*64  (reused as AGG2: N*16)
    float* H2       = H1;
    float* AGG2     = AGG1;

    const int B = 256;

    // 1) degrees -> norms (in place)
    k_zero<<<512, B, 0, stream>>>(norm_src, 2 * N);
    k_degrees<<<(E + B - 1) / B, B, 0, stream>>>(src, dst, norm_src, norm_dst, E);
    k_rsqrt<<<(2 * N + B - 1) / B, B, 0, stream>>>(norm_src, 2 * N);

    // 2) layer 1
    k_gemm1<<<(N + 31) / 32, B, 0, stream>>>(emb, W1, norm_src, H1, N);
    k_zero<<<2048, B, 0, stream>>>(AGG1, N * 64);
    {
        long threads = (long)E * 16;
        k_scatter1<<<(unsigned)((threads + B - 1) / B), B, 0, stream>>>(H1, src, dst, AGG1, E);
    }

    // 3) layer 2 (bias+ReLU+norms fused into GEMM2 A-stage)
    k_gemm2<<<(N + 127) / 128, B, 0, stream>>>(AGG1, W2, b1, norm_src, norm_dst, H2, N);
    k_zero<<<2048, B, 0, stream>>>(AGG2, N * 16);   // after GEMM2: AGG2 aliases AGG1
    {
        long threads = (long)E * 4;
        k_scatter2<<<(unsigned)((threads + B - 1) / B), B, 0, stream>>>(H2, src, dst, AGG2, E);
    }

    // 4) finalize into d_out
    k_finalize<<<(N * 16 + B - 1) / B, B, 0, stream>>>(AGG2, norm_dst, b2, out, N * 16);
}